// DNCLSTMEncoder_11630771438091
// MI455X (gfx1250) — compile-verified
//
#include <hip/hip_runtime.h>
#include <math.h>

#define SEQL  512
#define BATCH 64
#define EMB   512
#define HID   1024
#define RR    4
#define NW    25
#define WS    6
#define IFACE 47
#define INROW (EMB + RR*WS)   // 536, W_ih row length
#define G4    (4*HID)         // 4096
#define EPS   1e-6f

typedef __attribute__((ext_vector_type(2))) float v2f;
typedef __attribute__((ext_vector_type(8))) float v8f;

// D = A(16x4 f32) * B(4x16 f32) + C(16x16 f32), full fp32 matrix pipe
static __device__ __forceinline__ v8f wmma4(v2f a, v2f b, v8f c) {
  return __builtin_amdgcn_wmma_f32_16x16x4_f32(
      /*neg_a=*/false, a, /*neg_b=*/false, b,
      /*c_mod=*/(short)0, c, /*reuse_a=*/false, /*reuse_b=*/false);
}

static __device__ __forceinline__ float sigmoidf_(float x) {
  return 1.f / (1.f + __expf(-x));
}
static __device__ __forceinline__ float softplusf_(float x) {
  return (x > 20.f) ? x : log1pf(__expf(x));
}

// ---------------------------------------------------------------------------
// Phase 1 (fully parallel): Xg[t*B+b][n] = emb[src[t,b]] . Wih[n][0:512] + bias[n]
// M = SEQ*B = 32768, N = 4096, K = 512.  Wave blocking: 64 M-rows x 16 N-cols
// x 4 gates -> 16 v8f accumulators; A loaded per 16-row subtile, B (weights)
// amortized 4x across the M subtiles.  emb (64MB) + Wih_x (8MB) are L2-resident.
// grid = 4096 blocks x 256 thr (8 independent tile-waves per block).
// ---------------------------------------------------------------------------
__global__ __launch_bounds__(256)
void xgate_precompute_kernel(const int*   __restrict__ src,   // SEQ x BATCH
                             const float* __restrict__ emb,   // VOCAB x EMB
                             const float* __restrict__ Wih,   // 4096 x 536
                             const float* __restrict__ bias,  // 4096
                             float*       __restrict__ Xg)    // 32768 x 4096
{
  const int wid  = blockIdx.x * 8 + (threadIdx.x >> 5);  // 0..32767
  const int lane = threadIdx.x & 31;
  const int mblk = wid >> 6;          // 0..511 : 64-row M block
  const int nt   = wid & 63;          // 0..63  : 16-col N tile (per gate)
  const int m0   = mblk * 64;
  const int n0   = nt * 16;
  const int ml   = lane & 15;
  const int k2   = (lane >> 4) * 2;

  v8f acc[4][4] = {};   // [m-subtile][gate]

  // A row pointers: 4 subtiles of 16 rows; token gather (flat m == t*64+b == src index)
  const float* arow[4];
#pragma unroll
  for (int i = 0; i < 4; ++i)
    arow[i] = emb + (size_t)src[m0 + 16*i + ml] * EMB;

  const float* w0 = Wih + (size_t)(0*HID + n0 + ml) * INROW;
  const float* w1 = Wih + (size_t)(1*HID + n0 + ml) * INROW;
  const float* w2 = Wih + (size_t)(2*HID + n0 + ml) * INROW;
  const float* w3 = Wih + (size_t)(3*HID + n0 + ml) * INROW;

#pragma unroll 2
  for (int k = 0; k < EMB; k += 4) {
    v2f b0 = *(const v2f*)(w0 + k + k2);
    v2f b1 = *(const v2f*)(w1 + k + k2);
    v2f b2 = *(const v2f*)(w2 + k + k2);
    v2f b3 = *(const v2f*)(w3 + k + k2);
#pragma unroll
    for (int i = 0; i < 4; ++i) {
      v2f a = *(const v2f*)(arow[i] + k + k2);
      acc[i][0] = wmma4(a, b0, acc[i][0]);
      acc[i][1] = wmma4(a, b1, acc[i][1]);
      acc[i][2] = wmma4(a, b2, acc[i][2]);
      acc[i][3] = wmma4(a, b3, acc[i][3]);
    }
  }

  const int nIdx = n0 + ml;
  const int mhi  = (lane >> 4) * 8;
  const float bv[4] = { bias[0*HID + nIdx], bias[1*HID + nIdx],
                        bias[2*HID + nIdx], bias[3*HID + nIdx] };
#pragma unroll
  for (int i = 0; i < 4; ++i) {
#pragma unroll
    for (int g = 0; g < 4; ++g) {
#pragma unroll
      for (int p = 0; p < 8; ++p) {
        const int m = m0 + 16*i + p + mhi;
        Xg[(size_t)m * G4 + g*HID + nIdx] = acc[i][g][p] + bv[g];
      }
    }
  }
}

// ---------------------------------------------------------------------------
// One LSTM timestep.  If Xg != nullptr the x-part (and bias) was precomputed;
// the sequential K is then 1024 (h) + 24 (r).  Fallback path folds
// emb-gather x W_ih_x into the K loop.  grid = 64 blocks x 128 thr (4 waves);
// each wave owns a 16x16 tile for all 4 gates (i,f,g,o) so A loads once and
// the LSTM nonlinearity runs in-register.
// ---------------------------------------------------------------------------
__global__ __launch_bounds__(128)
void lstm_step_kernel(int t,
                      const int*   __restrict__ src,   // SEQ x BATCH
                      const float* __restrict__ emb,   // VOCAB x EMB
                      const float* __restrict__ Wih,   // 4096 x 536
                      const float* __restrict__ Whh,   // 4096 x 1024
                      const float* __restrict__ bias,  // 4096
                      const float* __restrict__ Xg,    // 32768 x 4096 or null
                      const float* __restrict__ hprev, // 64 x 1024
                      float*       __restrict__ hnext, // 64 x 1024
                      float*       __restrict__ cst,   // 64 x 1024
                      const float* __restrict__ rvec,  // 64 x 24
                      float*       __restrict__ out)   // SEQ x 64 x 1024
{
  const int wave = threadIdx.x >> 5;   // m-tile 0..3
  const int lane = threadIdx.x & 31;
  const int nt   = blockIdx.x;         // n-tile 0..63
  const int m0   = wave * 16;
  const int n0   = nt * 16;
  const int ml   = lane & 15;          // A: M row / B: N col for this lane
  const int k2   = (lane >> 4) * 2;    // lane K sub-offset {0,2}

  v8f acc0 = {}, acc1 = {}, acc2 = {}, acc3 = {};

  const float* wi0 = Wih + (size_t)(0*HID + n0 + ml) * INROW;
  const float* wi1 = Wih + (size_t)(1*HID + n0 + ml) * INROW;
  const float* wi2 = Wih + (size_t)(2*HID + n0 + ml) * INROW;
  const float* wi3 = Wih + (size_t)(3*HID + n0 + ml) * INROW;

  // ---- x-part (fallback only): K = 0..511, A rows via embedding gather ----
  if (Xg == nullptr) {
    const int token = src[t * BATCH + m0 + ml];
    const float* ax = emb + (size_t)token * EMB;
#pragma unroll 4
    for (int k = 0; k < EMB; k += 4) {
      v2f a  = *(const v2f*)(ax  + k + k2);
      v2f b0 = *(const v2f*)(wi0 + k + k2);
      v2f b1 = *(const v2f*)(wi1 + k + k2);
      v2f b2 = *(const v2f*)(wi2 + k + k2);
      v2f b3 = *(const v2f*)(wi3 + k + k2);
      acc0 = wmma4(a, b0, acc0);
      acc1 = wmma4(a, b1, acc1);
      acc2 = wmma4(a, b2, acc2);
      acc3 = wmma4(a, b3, acc3);
    }
  }

  // ---- h-part: K = 0..1023 (recurrent GEMM, weights L2-resident) ----
  const float* ah = hprev + (size_t)(m0 + ml) * HID;
  const float* wh0 = Whh + (size_t)(0*HID + n0 + ml) * HID;
  const float* wh1 = Whh + (size_t)(1*HID + n0 + ml) * HID;
  const float* wh2 = Whh + (size_t)(2*HID + n0 + ml) * HID;
  const float* wh3 = Whh + (size_t)(3*HID + n0 + ml) * HID;
#pragma unroll 4
  for (int k = 0; k < HID; k += 4) {
    v2f a  = *(const v2f*)(ah  + k + k2);
    v2f b0 = *(const v2f*)(wh0 + k + k2);
    v2f b1 = *(const v2f*)(wh1 + k + k2);
    v2f b2 = *(const v2f*)(wh2 + k + k2);
    v2f b3 = *(const v2f*)(wh3 + k + k2);
    acc0 = wmma4(a, b0, acc0);
    acc1 = wmma4(a, b1, acc1);
    acc2 = wmma4(a, b2, acc2);
    acc3 = wmma4(a, b3, acc3);
  }

  // ---- r-part: K = 0..23, W_ih columns [512, 536) ----
  const float* ar = rvec + (size_t)(m0 + ml) * (RR * WS);
#pragma unroll
  for (int k = 0; k < RR * WS; k += 4) {
    v2f a  = *(const v2f*)(ar  + k + k2);
    v2f b0 = *(const v2f*)(wi0 + EMB + k + k2);
    v2f b1 = *(const v2f*)(wi1 + EMB + k + k2);
    v2f b2 = *(const v2f*)(wi2 + EMB + k + k2);
    v2f b3 = *(const v2f*)(wi3 + EMB + k + k2);
    acc0 = wmma4(a, b0, acc0);
    acc1 = wmma4(a, b1, acc1);
    acc2 = wmma4(a, b2, acc2);
    acc3 = wmma4(a, b3, acc3);
  }

  // ---- epilogue: LSTM elementwise on the 16x16 tile ----
  const int nIdx = n0 + ml;
  float bi, bf, bg, bo;
  if (Xg == nullptr) {
    bi = bias[0*HID + nIdx]; bf = bias[1*HID + nIdx];
    bg = bias[2*HID + nIdx]; bo = bias[3*HID + nIdx];
  }
  const int mhi = (lane >> 4) * 8;   // D layout: lanes 16-31 hold M = p + 8
#pragma unroll
  for (int p = 0; p < 8; ++p) {
    const int m = m0 + p + mhi;
    float gi = acc0[p], gf = acc1[p], gg_ = acc2[p], go = acc3[p];
    if (Xg != nullptr) {
      const float* xg = Xg + ((size_t)t * BATCH + m) * G4 + nIdx;
      gi += xg[0*HID]; gf += xg[1*HID]; gg_ += xg[2*HID]; go += xg[3*HID];
    } else {
      gi += bi; gf += bf; gg_ += bg; go += bo;
    }
    const float ig = sigmoidf_(gi);
    const float fg = sigmoidf_(gf);
    const float gg = tanhf(gg_);
    const float og = sigmoidf_(go);
    const size_t idx = (size_t)m * HID + nIdx;
    const float cp = cst[idx];
    const float cn = fg * cp + ig * gg;
    cst[idx] = cn;
    const float hv = og * tanhf(cn);
    hnext[idx] = hv;
    out[(size_t)t * BATCH * HID + idx] = hv;
  }
}

// ---------------------------------------------------------------------------
// iface = h @ W_if^T + b_if (47 dots of 1024) + full DNC memory update.
// grid = 64 blocks (one per batch element), 256 threads (8 waves).
// ---------------------------------------------------------------------------
__global__ __launch_bounds__(256)
void iface_dnc_kernel(const float* __restrict__ h,    // 64 x 1024
                      const float* __restrict__ Wif,  // 47 x 1024
                      const float* __restrict__ bif,  // 47
                      float*       __restrict__ rvec, // 64 x 24
                      float*       __restrict__ Mg)   // 64 x 25 x 6
{
  const int b    = blockIdx.x;
  const int tid  = threadIdx.x;
  const int wave = tid >> 5;
  const int lane = tid & 31;

  __shared__ float sif[IFACE];
  __shared__ float sM[NW * WS];
  __shared__ float snorm[NW];
  __shared__ float sscore[NW];
  __shared__ float swr[RR][NW];
  __shared__ float serase[WS];
  __shared__ float srb[RR];
  __shared__ float srkn[RR];
  __shared__ float sscal[4];   // [0]=wkey norm [1]=write beta [2]=mx [3]=sum

  // --- iface dot products: one wave per output, lane-strided K, shfl reduce ---
  const float* hb = h + (size_t)b * HID;
  for (int j = wave; j < IFACE; j += 8) {
    const float* wr = Wif + (size_t)j * HID;
    float s = 0.f;
    for (int k = lane; k < HID; k += 32) s += hb[k] * wr[k];
#pragma unroll
    for (int off = 16; off >= 1; off >>= 1) s += __shfl_xor(s, off, 32);
    if (lane == 0) sif[j] = s + bif[j];
  }
  if (tid < NW * WS) sM[tid] = Mg[(size_t)b * NW * WS + tid];
  __syncthreads();

  // iface offsets: rk 0..23 | rbeta 24..27 | wkey 28..33 | wbeta 34
  //                erase 35..40 | wvec 41..46
  if (tid < NW) {
    float s = 0.f;
    for (int w = 0; w < WS; ++w) { float v = sM[tid*WS + w]; s += v*v; }
    snorm[tid] = sqrtf(s);
  }
  if (tid == 32) {
    float s = 0.f;
    for (int w = 0; w < WS; ++w) { float v = sif[28 + w]; s += v*v; }
    sscal[0] = sqrtf(s);
    sscal[1] = 1.f + softplusf_(sif[34]);
  }
  if (tid >= 64 && tid < 64 + WS) serase[tid - 64] = sigmoidf_(sif[35 + (tid - 64)]);
  if (tid >= 96 && tid < 96 + RR) srb[tid - 96] = 1.f + softplusf_(sif[24 + (tid - 96)]);
  __syncthreads();

  // write-content addressing
  if (tid < NW) {
    float d = 0.f;
    for (int w = 0; w < WS; ++w) d += sM[tid*WS + w] * sif[28 + w];
    sscore[tid] = sscal[1] * d / ((snorm[tid] + EPS) * (sscal[0] + EPS));
  }
  __syncthreads();
  if (tid == 0) {
    float mx = -1e30f;
    for (int n = 0; n < NW; ++n) mx = fmaxf(mx, sscore[n]);
    float s = 0.f;
    for (int n = 0; n < NW; ++n) s += __expf(sscore[n] - mx);
    sscal[2] = mx; sscal[3] = s;
  }
  __syncthreads();
  if (tid < NW) sscore[tid] = __expf(sscore[tid] - sscal[2]) / sscal[3]; // w_w
  __syncthreads();

  // memory write: M = M*(1 - w_w*erase) + w_w*write_vec
  if (tid < NW * WS) {
    const int n = tid / WS, w = tid % WS;
    const float m = sM[tid] * (1.f - sscore[n] * serase[w]) + sscore[n] * sif[41 + w];
    sM[tid] = m;
    Mg[(size_t)b * NW * WS + tid] = m;
  }
  __syncthreads();

  // re-normalize rows of new M; read-key norms
  if (tid < NW) {
    float s = 0.f;
    for (int w = 0; w < WS; ++w) { float v = sM[tid*WS + w]; s += v*v; }
    snorm[tid] = sqrtf(s);
  }
  if (tid >= 32 && tid < 32 + RR) {
    const int r = tid - 32;
    float s = 0.f;
    for (int w = 0; w < WS; ++w) { float v = sif[r*WS + w]; s += v*v; }
    srkn[r] = sqrtf(s);
  }
  __syncthreads();

  // read-content addressing (4 x 25 scores)
  if (tid < RR * NW) {
    const int r = tid / NW, n = tid % NW;
    float d = 0.f;
    for (int w = 0; w < WS; ++w) d += sM[n*WS + w] * sif[r*WS + w];
    swr[r][n] = srb[r] * d / ((snorm[n] + EPS) * (srkn[r] + EPS));
  }
  __syncthreads();
  if (tid < RR) {
    float mx = -1e30f;
    for (int n = 0; n < NW; ++n) mx = fmaxf(mx, swr[tid][n]);
    float s = 0.f;
    for (int n = 0; n < NW; ++n) { float e = __expf(swr[tid][n] - mx); swr[tid][n] = e; s += e; }
    const float inv = 1.f / s;
    for (int n = 0; n < NW; ++n) swr[tid][n] *= inv;
  }
  __syncthreads();

  // read vectors r[r][w] = sum_n w_r[r][n] * M[n][w]
  if (tid < RR * WS) {
    const int r = tid / WS, w = tid % WS;
    float s = 0.f;
    for (int n = 0; n < NW; ++n) s += swr[r][n] * sM[n*WS + w];
    rvec[(size_t)b * (RR*WS) + tid] = s;
  }
}

// ---------------------------------------------------------------------------
// lengths = min(sum(src != pad, axis=0), SEQ-1); last_state = outputs[len, b]
// ---------------------------------------------------------------------------
__global__ __launch_bounds__(256)
void gather_last_kernel(const int*   __restrict__ src,
                        const int*   __restrict__ pad_ptr,
                        const float* __restrict__ out,
                        float*       __restrict__ last)
{
  __shared__ int sred[256];
  const int b = blockIdx.x, tid = threadIdx.x;
  const int pad = *pad_ptr;
  int cnt = 0;
  for (int t = tid; t < SEQL; t += 256) cnt += (src[t * BATCH + b] != pad);
  sred[tid] = cnt;
  __syncthreads();
  for (int off = 128; off >= 1; off >>= 1) {
    if (tid < off) sred[tid] += sred[tid + off];
    __syncthreads();
  }
  const int len = min(sred[0], SEQL - 1);
  for (int k = tid; k < HID; k += 256)
    last[(size_t)b * HID + k] = out[(size_t)len * BATCH * HID + (size_t)b * HID + k];
}

extern "C" void kernel_launch(void* const* d_in, const int* in_sizes, int n_in,
                              void* d_out, int out_size, void* d_ws, size_t ws_size,
                              hipStream_t stream) {
  const int*   src  = (const int*)  d_in[0];
  const int*   padp = (const int*)  d_in[1];
  const float* emb  = (const float*)d_in[2];
  const float* Wih  = (const float*)d_in[3];
  const float* Whh  = (const float*)d_in[4];
  const float* bias = (const float*)d_in[5];
  const float* Wif  = (const float*)d_in[6];
  const float* bif  = (const float*)d_in[7];

  float* out  = (float*)d_out;                         // SEQ*BATCH*HID
  float* last = out + (size_t)SEQL * BATCH * HID;      // BATCH*HID

  const size_t stateFloats = (size_t)3 * BATCH * HID + BATCH * RR * WS + BATCH * NW * WS;
  const size_t xgFloats    = (size_t)SEQL * BATCH * G4;   // 512 MB

  float* ws = (float*)d_ws;
  float* h0 = ws;                          // 64*1024
  float* h1 = h0 + BATCH * HID;            // 64*1024
  float* cs = h1 + BATCH * HID;            // 64*1024
  float* rv = cs + BATCH * HID;            // 64*24
  float* Mg = rv + BATCH * RR * WS;        // 64*25*6
  float* Xg = Mg + BATCH * NW * WS;        // optional 32768*4096

  const bool useXg = ws_size >= (stateFloats + xgFloats) * sizeof(float);

  hipMemsetAsync(h0, 0, (size_t)BATCH * HID * sizeof(float), stream);
  hipMemsetAsync(cs, 0, (size_t)BATCH * HID * sizeof(float), stream);
  hipMemsetAsync(rv, 0, (size_t)BATCH * RR * WS * sizeof(float), stream);
  hipMemsetAsync(Mg, 0, (size_t)BATCH * NW * WS * sizeof(float), stream);

  if (useXg) {
    // 32768 tile-waves; 8 per 256-thread block -> 4096 blocks, saturates the chip
    xgate_precompute_kernel<<<4096, 256, 0, stream>>>(src, emb, Wih, bias, Xg);
  }

  const float* xgArg = useXg ? Xg : nullptr;
  for (int t = 0; t < SEQL; ++t) {
    float* hp = (t & 1) ? h1 : h0;
    float* hn = (t & 1) ? h0 : h1;
    lstm_step_kernel<<<64, 128, 0, stream>>>(t, src, emb, Wih, Whh, bias,
                                             xgArg, hp, hn, cs, rv, out);
    iface_dnc_kernel<<<64, 256, 0, stream>>>(hn, Wif, bif, rv, Mg);
  }
  gather_last_kernel<<<64, 256, 0, stream>>>(src, padp, out, last);
}